// DataAdaptiveWeightEnsemblingModel_15109694947865
// MI455X (gfx1250) — compile-verified
//
#include <hip/hip_runtime.h>
#include <hip/hip_bf16.h>

// B=16, D=1024, DFF=4096, E=16, L=4
// out = per-sample expert-merged 2-layer MLP; coefficients folded into WMMA A-operand.
// HBM-bound (~550MB streamed @ 23.3 TB/s => ~24us floor); fp32 WMMA keeps full
// precision while using the matrix pipe. Weights streamed with NT hint to keep
// L2 for the reused operands (x, h1, codings, partials).

typedef __attribute__((ext_vector_type(2))) float v2f;
typedef __attribute__((ext_vector_type(8))) float v8f;

#define NB   16
#define ND   1024
#define NDFF 4096
#define NE   16

// ---------------- gate ----------------
__global__ void gate1_kernel(const float* __restrict__ x,
                             const float* __restrict__ gW1,
                             const float* __restrict__ gb1,
                             float* __restrict__ h) {
  int t = blockIdx.x * blockDim.x + threadIdx.x;   // over B*D = 16384
  int b = t >> 10;
  int j = t & 1023;
  const float* xr = x + b * ND;
  const float* wr = gW1 + (size_t)j * ND;
  float s = gb1[j];
  for (int d = 0; d < ND; ++d) s += xr[d] * wr[d];
  h[t] = fmaxf(s, 0.0f);
}

__global__ void gate2_kernel(const float* __restrict__ h,
                             const float* __restrict__ gW2,
                             const float* __restrict__ gb2,
                             float* __restrict__ cod) {
  int t = blockIdx.x * blockDim.x + threadIdx.x;   // over B*64 = 1024
  if (t >= NB * 64) return;
  int b = t >> 6;
  int o = t & 63;
  const float* hr = h + b * ND;
  const float* wr = gW2 + (size_t)o * ND;
  float s = gb2[o];
  for (int j = 0; j < ND; ++j) s += hr[j] * wr[j];
  cod[t] = s;   // cod[b*64 + e*4 + slot]
}

// ---------------- WMMA partial GEMM ----------------
// Unit u = ((n*17 + m)*KC + kc): tile n (16 output cols), matrix m (0=base,
// 1..16=expert m-1, with coefficient folded into A), K-chunk kc (span Ktot/KC=512).
// Computes 16(batch) x 16(cols) fp32 partial via V_WMMA_F32_16X16X4_F32, K=512.
__global__ __launch_bounds__(32) void gemm_partial_kernel(
    const float* __restrict__ A,     // [16, Ktot]
    const float* __restrict__ Wb,    // [Ntot, Ktot] base
    const float* __restrict__ Wtv,   // [E, Ntot, Ktot] task vectors
    const float* __restrict__ cod,   // [B, E, L]
    float* __restrict__ P,           // [gridDim.x, 256] partials
    int Ktot, int KC, int slot, int Ntot) {
  const int lane = threadIdx.x;
  const int u  = blockIdx.x;
  const int kc = u % KC;
  const int m  = (u / KC) % 17;
  const int n  = u / (KC * 17);
  const int kspan = Ktot / KC;           // 512
  const int row = lane & 15;             // A: M row / B: N col
  const int hi  = lane >> 4;             // K sub-phase
  const int k0  = kc * kspan;

  const float* Wsrc = (m == 0) ? Wb : (Wtv + (size_t)(m - 1) * Ntot * Ktot);
  const v2f* aP = (const v2f*)(A + (size_t)row * Ktot + k0) + hi;
  const v2f* bP = (const v2f*)(Wsrc + (size_t)(n * 16 + row) * Ktot + k0) + hi;
  const float s = (m == 0) ? 1.0f : cod[(row * NE + (m - 1)) * 4 + slot];

  v8f acc = {0.f, 0.f, 0.f, 0.f, 0.f, 0.f, 0.f, 0.f};
  const int chunks = kspan >> 2;         // 128 wmma per unit
  for (int kk = 0; kk < chunks; kk += 8) {
    v2f av[8], bv[8];
#pragma unroll
    for (int i = 0; i < 8; ++i) {
      av[i] = aP[(kk + i) * 2];                                  // reused (x/h1): keep temporal
      bv[i] = __builtin_nontemporal_load(bP + (kk + i) * 2);     // streamed weights: TH=NT
    }
#pragma unroll
    for (int i = 0; i < 8; ++i) {
      v2f a = s * av[i];
      acc = __builtin_amdgcn_wmma_f32_16x16x4_f32(
          false, a, false, bv[i], (short)0, acc, false, false);
    }
  }
  float* Pp = P + (size_t)u * 256 + lane;
#pragma unroll
  for (int r = 0; r < 8; ++r) Pp[r * 32] = acc[r];   // element (r,lane) = (M=r+8*(lane/16), N=lane%16)
}

// ---------------- reduction + bias-tv + activation ----------------
__global__ void reduce_apply_kernel(const float* __restrict__ P,
                                    const float* __restrict__ bias,
                                    const float* __restrict__ tvb,   // [E, Ntot]
                                    const float* __restrict__ cod,
                                    float* __restrict__ Out,         // [B, Ntot]
                                    int Ntot, int KC, int slotB, int do_relu) {
  int t = blockIdx.x * blockDim.x + threadIdx.x;   // over B*Ntot
  if (t >= NB * Ntot) return;
  int b = t / Ntot;
  int f = t - b * Ntot;
  int n = f >> 4, col = f & 15;
  int lane = col + ((b >> 3) << 4);
  int idx = (b & 7) * 32 + lane;
  float sum = bias[f];
  int ubase = n * 17 * KC;
  for (int mk = 0; mk < 17 * KC; ++mk)
    sum += P[(size_t)(ubase + mk) * 256 + idx];
  for (int e = 0; e < NE; ++e)
    sum += cod[(b * NE + e) * 4 + slotB] * tvb[(size_t)e * Ntot + f];
  if (do_relu) sum = fmaxf(sum, 0.0f);
  Out[t] = sum;
}

extern "C" void kernel_launch(void* const* d_in, const int* in_sizes, int n_in,
                              void* d_out, int out_size, void* d_ws, size_t ws_size,
                              hipStream_t stream) {
  const float* x    = (const float*)d_in[0];
  const float* gW1  = (const float*)d_in[1];
  const float* gb1  = (const float*)d_in[2];
  const float* gW2  = (const float*)d_in[3];
  const float* gb2  = (const float*)d_in[4];
  const float* bW1  = (const float*)d_in[5];
  const float* bb1  = (const float*)d_in[6];
  const float* bW2  = (const float*)d_in[7];
  const float* bb2  = (const float*)d_in[8];
  const float* tvW1 = (const float*)d_in[9];
  const float* tvb1 = (const float*)d_in[10];
  const float* tvW2 = (const float*)d_in[11];
  const float* tvb2 = (const float*)d_in[12];
  float* out = (float*)d_out;

  float* ws  = (float*)d_ws;
  float* h    = ws;                       // 16*1024
  float* cod  = h + NB * ND;              // 16*64
  float* h1   = cod + NB * 64;            // 16*4096
  float* P    = h1 + NB * NDFF;           // 8704*256 partials (reused by both layers)

  // gate
  gate1_kernel<<<(NB * ND) / 256, 256, 0, stream>>>(x, gW1, gb1, h);
  gate2_kernel<<<4, 256, 0, stream>>>(h, gW2, gb2, cod);

  // layer 1: N tiles = 4096/16 = 256, KC=2 -> 256*17*2 = 8704 wave-units
  gemm_partial_kernel<<<256 * 17 * 2, 32, 0, stream>>>(
      x, bW1, tvW1, cod, P, ND, 2, /*slot W1*/ 0, NDFF);
  reduce_apply_kernel<<<(NB * NDFF) / 256, 256, 0, stream>>>(
      P, bb1, tvb1, cod, h1, NDFF, 2, /*slot b1*/ 1, /*relu*/ 1);

  // layer 2: N tiles = 1024/16 = 64, KC=8 -> 64*17*8 = 8704 wave-units
  gemm_partial_kernel<<<64 * 17 * 8, 32, 0, stream>>>(
      h1, bW2, tvW2, cod, P, NDFF, 8, /*slot W2*/ 2, ND);
  reduce_apply_kernel<<<(NB * ND) / 256, 256, 0, stream>>>(
      P, bb2, tvb2, cod, out, ND, 8, /*slot b2*/ 3, /*relu*/ 0);
}